// GIoULoss_33672543601399
// MI455X (gfx1250) — compile-verified
//
#include <hip/hip_runtime.h>

typedef __attribute__((ext_vector_type(2))) float v2f;
typedef __attribute__((ext_vector_type(4))) float v4f;
typedef __attribute__((ext_vector_type(8))) float v8f;

// Full-block sum reduction. Requires: all threads in the block reach this
// point with no divergence (EXEC must be all-1s for WMMA). Valid result only
// on threadIdx.x == 0.
__device__ __forceinline__ float block_reduce_sum(float s, float* lds) {
    // Stage 1: 32 lanes -> 16 partial sums with one f32 WMMA.
    //   A (16x4) = all ones, B (4x16) carries lane partials in b.x, zeros in
    //   b.y. Regardless of which K-rows b.x maps to, the zero rows guarantee
    //   D[m][n] = s(lane n) + s(lane n+16) for every m -> exact f32 math.
    v2f a; a.x = 1.0f; a.y = 1.0f;
    v2f b; b.x = s;    b.y = 0.0f;
    v8f c = {};
    c = __builtin_amdgcn_wmma_f32_16x16x4_f32(
        /*neg_a=*/false, a, /*neg_b=*/false, b,
        /*c_mod=*/(short)0, c, /*reuse_a=*/false, /*reuse_b=*/false);
    float d = c[0];  // D row 0 (lanes 0-15) / row 8 (lanes 16-31): colsum[n]

    // Stage 2: 16 -> 1 butterfly within each 16-lane half (both halves hold
    // the same colsum vector, so every lane ends with the full wave sum).
    d += __shfl_xor(d, 1, 32);
    d += __shfl_xor(d, 2, 32);
    d += __shfl_xor(d, 4, 32);
    d += __shfl_xor(d, 8, 32);

    // Stage 3: combine the (up to 8) wave sums via LDS.
    const int lane = threadIdx.x & 31;
    const int wave = threadIdx.x >> 5;
    if (lane == 0) lds[wave] = d;
    __syncthreads();
    float total = 0.0f;
    if (threadIdx.x == 0) {
        const int nwaves = blockDim.x >> 5;
        for (int w = 0; w < nwaves; ++w) total += lds[w];
    }
    return total;
}

__global__ void giou_partial_kernel(const v4f* __restrict__ pred,
                                    const v4f* __restrict__ tgt,
                                    float* __restrict__ partials, int n) {
    __shared__ float lds[8];
    const int tid    = blockIdx.x * blockDim.x + threadIdx.x;
    const int stride = gridDim.x * blockDim.x;

    float sum = 0.0f;
    for (int i = tid; i < n; i += stride) {
        // Streaming 16B loads with non-temporal hint: data is single-use and
        // larger than L2, so bypass regular-temporal residency.
        v4f p = __builtin_nontemporal_load(&pred[i]);
        v4f t = __builtin_nontemporal_load(&tgt[i]);

        const float area_p = (p.z - p.x) * (p.w - p.y);
        const float area_t = (t.z - t.x) * (t.w - t.y);

        const float iw = fmaxf(fminf(p.z, t.z) - fmaxf(p.x, t.x), 0.0f);
        const float ih = fmaxf(fminf(p.w, t.w) - fmaxf(p.y, t.y), 0.0f);
        const float inter = iw * ih;
        const float uni   = area_p + area_t - inter;
        const float iou   = inter / uni;

        const float ew = fmaxf(fmaxf(p.z, t.z) - fminf(p.x, t.x), 0.0f);
        const float eh = fmaxf(fmaxf(p.w, t.w) - fminf(p.y, t.y), 0.0f);
        const float enc = ew * eh;

        const float giou = iou - (enc - uni) / enc;
        sum += 1.0f - giou;
    }

    const float total = block_reduce_sum(sum, lds);
    if (threadIdx.x == 0) partials[blockIdx.x] = total;
}

__global__ void reduce_partials_kernel(const float* __restrict__ partials,
                                       float* __restrict__ out,
                                       int nparts, float inv_n) {
    __shared__ float lds[8];
    float sum = 0.0f;
    for (int i = threadIdx.x; i < nparts; i += blockDim.x) sum += partials[i];
    const float total = block_reduce_sum(sum, lds);
    if (threadIdx.x == 0) out[0] = total * inv_n;
}

extern "C" void kernel_launch(void* const* d_in, const int* in_sizes, int n_in,
                              void* d_out, int out_size, void* d_ws, size_t ws_size,
                              hipStream_t stream) {
    const v4f* pred = (const v4f*)d_in[0];
    const v4f* tgt  = (const v4f*)d_in[1];
    const int n = in_sizes[0] / 4;   // flat float count -> box count

    float* partials = (float*)d_ws;  // nblocks floats of scratch

    const int threads = 256;         // 8 wave32s per block
    int blocks = 2048;               // 512K threads: ~16 boxes/lane, deep MLP
    const int maxb = (n + threads - 1) / threads;
    if (blocks > maxb) blocks = maxb;
    if (blocks < 1) blocks = 1;

    giou_partial_kernel<<<blocks, threads, 0, stream>>>(pred, tgt, partials, n);
    reduce_partials_kernel<<<1, threads, 0, stream>>>(
        partials, (float*)d_out, blocks, 1.0f / (float)n);
}